// Attn_47072841564500
// MI455X (gfx1250) — compile-verified
//
#include <hip/hip_runtime.h>

// Shapes from the reference
#define HID   1024
#define SEQ   2048
#define BATCH 64

typedef __attribute__((ext_vector_type(2))) float v2f;
typedef __attribute__((ext_vector_type(8))) float v8f;

// ---------------------------------------------------------------------------
// Kernel 1: v[b,m] = sum_n h[b,n] * W[n,m]   (64x1024 = 64x1024 @ 1024x1024)
// One wave32 per 16x16 output tile, v_wmma_f32_16x16x4_f32 over K in steps of 4.
// A layout (16x4 f32): lanes 0-15 hold M=lane K={0,1}; lanes 16-31 M=lane-16 K={2,3}
// B layout (4x16 f32): VGPR r = row K=r (lanes 0-15, N=lane) / K=r+2 (lanes 16-31)
// D layout (16x16 f32): VGPR r = M=r (lanes 0-15) / M=r+8 (lanes 16-31), N=lane%16
// ---------------------------------------------------------------------------
__global__ __launch_bounds__(32)
void vproj_wmma(const float* __restrict__ h, const float* __restrict__ W,
                float* __restrict__ v) {
    const int nBase = blockIdx.x * 16;
    const int mBase = blockIdx.y * 16;
    const int lane  = threadIdx.x;
    const int hi    = lane >> 4;      // which half of the wave
    const int lrow  = lane & 15;

    v8f acc = {};
    const int hRow = (mBase + lrow) * HID;
    for (int k = 0; k < HID; k += 4) {
        const int ka = k + 2 * hi;            // K index pair base for this half
        v2f a;
        a.x = h[hRow + ka];                   // A: M=lrow, K=ka
        a.y = h[hRow + ka + 1];               // A: M=lrow, K=ka+1
        v2f bm;
        bm.x = W[(size_t)(ka)     * HID + nBase + lrow];  // B: K=ka,   N=lrow
        bm.y = W[(size_t)(ka + 1) * HID + nBase + lrow];  // B: K=ka+1, N=lrow
        acc = __builtin_amdgcn_wmma_f32_16x16x4_f32(
            /*neg_a=*/false, a, /*neg_b=*/false, bm,
            /*c_mod=*/(short)0, acc, /*reuse_a=*/false, /*reuse_b=*/false);
    }
#pragma unroll
    for (int r = 0; r < 8; ++r)
        v[(size_t)(mBase + r + 8 * hi) * HID + nBase + lrow] = acc[r];
}

// ---------------------------------------------------------------------------
// Kernel 2 (bandwidth-dominant, 512 MB stream):
//   scores[b,s] = enc[s,b,:] . v[b,:]
// Each block owns one b, stages v[b] (4 KB) in LDS, 8 waves x 16 rows each.
// Per row: 32 lanes x 8 float4 loads (4 KB contiguous) + FMA + shuffle reduce.
// ---------------------------------------------------------------------------
#define ROWS_PER_BLOCK 128
__global__ __launch_bounds__(256)
void scores_kernel(const float* __restrict__ enc, const float* __restrict__ v,
                   float* __restrict__ scores) {
    __shared__ float vsh[HID];
    const int b     = blockIdx.y;
    const int sBase = blockIdx.x * ROWS_PER_BLOCK;
    const int t     = threadIdx.x;

    // stage v[b,:] into LDS: 256 threads x one float4 = 4 KB
    ((float4*)vsh)[t] = ((const float4*)(v + (size_t)b * HID))[t];
    __syncthreads();

    const int wave = t >> 5;
    const int lane = t & 31;
    const float4* vsh4 = (const float4*)vsh;

    for (int r = 0; r < ROWS_PER_BLOCK / 8; ++r) {
        const int s = sBase + wave + r * 8;
        const float4* row =
            (const float4*)(enc + ((size_t)s * BATCH + b) * HID);
        float4 acc = {0.f, 0.f, 0.f, 0.f};
#pragma unroll
        for (int i = 0; i < 8; ++i) {
            float4 e = row[lane + i * 32];   // 512 B contiguous per iter/wave
            float4 w = vsh4[lane + i * 32];
            acc.x += e.x * w.x;
            acc.y += e.y * w.y;
            acc.z += e.z * w.z;
            acc.w += e.w * w.w;
        }
        float sum = (acc.x + acc.y) + (acc.z + acc.w);
#pragma unroll
        for (int off = 16; off > 0; off >>= 1)
            sum += __shfl_xor(sum, off, 32);   // wave32 reduction
        if (lane == 0) scores[(size_t)b * SEQ + s] = sum;
    }
}

// ---------------------------------------------------------------------------
// Kernel 3: softmax over s (len 2048) per b. Bias term h.b cancels (shift
// invariance), so scores need no constant added. 64 blocks x 256 threads,
// 8 elements per thread, two-level (wave shuffle + LDS) reductions.
// ---------------------------------------------------------------------------
__global__ __launch_bounds__(256)
void softmax_kernel(const float* __restrict__ scores, float* __restrict__ out) {
    __shared__ float redmax[8];
    __shared__ float redsum[8];
    const int b    = blockIdx.x;
    const int t    = threadIdx.x;
    const int lane = t & 31;
    const int wave = t >> 5;

    float x[8];
    float mx = -3.402823466e38f;
#pragma unroll
    for (int k = 0; k < 8; ++k) {
        x[k] = scores[(size_t)b * SEQ + t + k * 256];
        mx = fmaxf(mx, x[k]);
    }
#pragma unroll
    for (int off = 16; off > 0; off >>= 1)
        mx = fmaxf(mx, __shfl_xor(mx, off, 32));
    if (lane == 0) redmax[wave] = mx;
    __syncthreads();
#pragma unroll
    for (int w = 0; w < 8; ++w) mx = fmaxf(mx, redmax[w]);

    float lsum = 0.f;
#pragma unroll
    for (int k = 0; k < 8; ++k) {
        x[k] = __expf(x[k] - mx);
        lsum += x[k];
    }
#pragma unroll
    for (int off = 16; off > 0; off >>= 1)
        lsum += __shfl_xor(lsum, off, 32);
    if (lane == 0) redsum[wave] = lsum;
    __syncthreads();
    float tot = 0.f;
#pragma unroll
    for (int w = 0; w < 8; ++w) tot += redsum[w];
    const float inv = 1.0f / tot;

#pragma unroll
    for (int k = 0; k < 8; ++k)
        out[(size_t)b * SEQ + t + k * 256] = x[k] * inv;   // out is [B,1,S]
}

// ---------------------------------------------------------------------------
extern "C" void kernel_launch(void* const* d_in, const int* in_sizes, int n_in,
                              void* d_out, int out_size, void* d_ws,
                              size_t ws_size, hipStream_t stream) {
    (void)in_sizes; (void)n_in; (void)out_size; (void)ws_size;
    const float* h   = (const float*)d_in[0];  // [1,64,1024]
    const float* enc = (const float*)d_in[1];  // [2048,64,1024]
    const float* W   = (const float*)d_in[2];  // [1024,1024]
    // d_in[3] (bias) is mathematically irrelevant: its contribution to every
    // score in a softmax row is the same constant h[b].bias, which cancels.
    float* out = (float*)d_out;

    float* v      = (float*)d_ws;          // 64*1024 f32  (256 KB)
    float* scores = v + BATCH * HID;       // 64*2048 f32  (512 KB)

    // 1) v = h @ W via fp32 WMMA (256 waves, one 16x16 tile each)
    vproj_wmma<<<dim3(HID / 16, BATCH / 16), 32, 0, stream>>>(h, W, v);
    // 2) bandwidth pass over enc (512 MB -> ~22 us at 23.3 TB/s)
    scores_kernel<<<dim3(SEQ / ROWS_PER_BLOCK, BATCH), 256, 0, stream>>>(
        enc, v, scores);
    // 3) softmax rows
    softmax_kernel<<<BATCH, 256, 0, stream>>>(scores, out);
}